// AttentionDecoderCell_78606491452475
// MI455X (gfx1250) — compile-verified
//
#include <hip/hip_runtime.h>
#include <hip/hip_bf16.h>

typedef __attribute__((ext_vector_type(2))) float v2f;
typedef __attribute__((ext_vector_type(8))) float v8f;

#define HD 256   // hidden
#define FD 64    // feature
#define SD 512   // encoder timesteps
#define KA 320   // H + F
#define G3 768   // 3*H

static __device__ __forceinline__ v8f wmma_f32(v2f a, v2f b, v8f c) {
    // V_WMMA_F32_16X16X4_F32: D = A(16x4) * B(4x16) + C(16x16), full fp32
    return __builtin_amdgcn_wmma_f32_16x16x4_f32(
        /*neg_a=*/false, a, /*neg_b=*/false, b,
        /*c_mod=*/(short)0, c, /*reuse_a=*/false, /*reuse_b=*/false);
}

// ---------------------------------------------------------------------------
// K1: logits[b,s] = attn_in[b,:] . attn_W[s,:]   (M=B, N=512, K=320)
// One block = 16 batch rows x all 512 columns. 8 waves; each wave owns 4
// 16x16 N-tiles (stride 128), sharing one A fragment per K-step.
// ---------------------------------------------------------------------------
__global__ __launch_bounds__(256) void k_attn_logits(
    const float* __restrict__ ph, const float* __restrict__ y,
    const float* __restrict__ attn_W, float* __restrict__ logits)
{
    __shared__ __align__(16) float As[16 * KA];      // 20 KB A tile
    const int b0 = blockIdx.x * 16;
    const int t  = threadIdx.x;
    for (int i = t; i < 16 * KA; i += 256) {
        int r = i / KA, k = i - r * KA;
        As[i] = (k < HD) ? ph[(b0 + r) * HD + k] : y[(b0 + r) * FD + (k - HD)];
    }
    __syncthreads();

    const int lane = t & 31, wave = t >> 5;
    const int m  = lane & 15;                 // A row
    const int kh = (lane >> 4) << 1;          // K sub-offset: 0 or 2
    const int nn = lane & 15;                 // B column within tile
    v8f acc[4] = {};

    for (int k = 0; k < KA; k += 4) {
        v2f a = *(const v2f*)(&As[m * KA + k + kh]);
#pragma unroll
        for (int s = 0; s < 4; ++s) {
            int n = wave * 16 + s * 128 + nn;
            v2f b = *(const v2f*)(&attn_W[n * KA + k + kh]);
            acc[s] = wmma_f32(a, b, acc[s]);
        }
    }

    const int rbase = (lane >> 4) << 3;       // lanes 16-31 hold M=8..15
#pragma unroll
    for (int s = 0; s < 4; ++s) {
        int n = wave * 16 + s * 128 + nn;
#pragma unroll
        for (int i = 0; i < 8; ++i)
            logits[(size_t)(b0 + rbase + i) * SD + n] = acc[s][i];
    }
}

// ---------------------------------------------------------------------------
// K2: row softmax over S=512 (adds attn_b here), in place.
// ---------------------------------------------------------------------------
__global__ __launch_bounds__(256) void k_softmax(
    float* __restrict__ logits, const float* __restrict__ attn_b)
{
    __shared__ float red[256];
    const int b = blockIdx.x, t = threadIdx.x;
    float v0 = logits[(size_t)b * SD + t]       + attn_b[t];
    float v1 = logits[(size_t)b * SD + 256 + t] + attn_b[256 + t];
    red[t] = fmaxf(v0, v1); __syncthreads();
    for (int o = 128; o > 0; o >>= 1) {
        if (t < o) red[t] = fmaxf(red[t], red[t + o]);
        __syncthreads();
    }
    float mx = red[0]; __syncthreads();
    float e0 = __expf(v0 - mx), e1 = __expf(v1 - mx);
    red[t] = e0 + e1; __syncthreads();
    for (int o = 128; o > 0; o >>= 1) {
        if (t < o) red[t] += red[t + o];
        __syncthreads();
    }
    float inv = 1.0f / red[0];
    logits[(size_t)b * SD + t]       = e0 * inv;
    logits[(size_t)b * SD + 256 + t] = e1 * inv;
}

// ---------------------------------------------------------------------------
// K3: ctx[b,h] = sum_s w[b,s] * enc[b,s,h].  The 1.07 GB stream: this kernel
// sets the runtime floor. float4 (b128) coalesced loads, weights in LDS,
// s-dimension split 4 ways across the block, prefetch ahead of the stream.
// ---------------------------------------------------------------------------
__global__ __launch_bounds__(256) void k_ctx(
    const float* __restrict__ enc, const float* __restrict__ attn_w,
    float* __restrict__ ctx)
{
    __shared__ float w[SD];
    __shared__ float4 part[256];
    const int b = blockIdx.x, t = threadIdx.x;
    w[t]       = attn_w[(size_t)b * SD + t];
    w[256 + t] = attn_w[(size_t)b * SD + 256 + t];
    __syncthreads();

    const int h4 = (t & 63) << 2;       // 64 lanes cover H=256 via float4
    const int sc = t >> 6;              // 4 s-chunks of 128
    const float* base = enc + (size_t)b * SD * HD + (size_t)sc * 128 * HD + h4;

    float4 acc = make_float4(0.f, 0.f, 0.f, 0.f);
    for (int s = 0; s < 128; ++s) {
        int sp = (s + 8 < 128) ? s + 8 : s;
        __builtin_prefetch(base + (size_t)sp * HD, 0, 1);   // global_prefetch_b8
        float ws = w[sc * 128 + s];
        float4 ev = *(const float4*)(base + (size_t)s * HD);
        acc.x += ws * ev.x; acc.y += ws * ev.y;
        acc.z += ws * ev.z; acc.w += ws * ev.w;
    }
    part[t] = acc; __syncthreads();
    if (t < 64) {
        float4 a = part[t], b1 = part[t + 64], c = part[t + 128], d = part[t + 192];
        float4 r;
        r.x = a.x + b1.x + c.x + d.x;
        r.y = a.y + b1.y + c.y + d.y;
        r.z = a.z + b1.z + c.z + d.z;
        r.w = a.w + b1.w + c.w + d.w;
        *(float4*)(ctx + (size_t)b * HD + (t << 2)) = r;
    }
}

// ---------------------------------------------------------------------------
// K4: gi = ctx @ W_ih^T, gh = prev_hidden @ W_hh^T  (M=B, N=768, K=256)
// Block tile: 16 batch rows x 128 gate columns; both A tiles staged in LDS,
// two fp32 WMMA accumulators per wave.
// ---------------------------------------------------------------------------
__global__ __launch_bounds__(256) void k_gru_gemm(
    const float* __restrict__ ctx, const float* __restrict__ ph,
    const float* __restrict__ W_ih, const float* __restrict__ W_hh,
    float* __restrict__ gi, float* __restrict__ gh)
{
    __shared__ __align__(16) float Ac[16 * HD];   // 16 KB
    __shared__ __align__(16) float Ah[16 * HD];   // 16 KB
    const int b0 = blockIdx.x * 16;
    const int g0 = blockIdx.y * 128;
    const int t  = threadIdx.x;
    for (int i = t; i < 16 * HD; i += 256) {
        int r = i >> 8, k = i & 255;
        Ac[i] = ctx[(size_t)(b0 + r) * HD + k];
        Ah[i] = ph [(size_t)(b0 + r) * HD + k];
    }
    __syncthreads();

    const int lane = t & 31, wave = t >> 5;
    const int m  = lane & 15;
    const int kh = (lane >> 4) << 1;
    const int n  = g0 + wave * 16 + (lane & 15);
    v8f ci = {}, ch = {};

    for (int k = 0; k < HD; k += 4) {
        v2f ac = *(const v2f*)(&Ac[m * HD + k + kh]);
        v2f ah = *(const v2f*)(&Ah[m * HD + k + kh]);
        v2f bi = *(const v2f*)(&W_ih[(size_t)n * HD + k + kh]);
        v2f bh = *(const v2f*)(&W_hh[(size_t)n * HD + k + kh]);
        ci = wmma_f32(ac, bi, ci);
        ch = wmma_f32(ah, bh, ch);
    }

    const int rbase = (lane >> 4) << 3;
#pragma unroll
    for (int i = 0; i < 8; ++i) {
        gi[(size_t)(b0 + rbase + i) * G3 + n] = ci[i];
        gh[(size_t)(b0 + rbase + i) * G3 + n] = ch[i];
    }
}

// ---------------------------------------------------------------------------
// K5: GRU gate nonlinearities, h_new, and out = h_new . out_W + out_b.
// One block per batch row, thread == h.
// ---------------------------------------------------------------------------
__global__ __launch_bounds__(256) void k_gates(
    const float* __restrict__ gi, const float* __restrict__ gh,
    const float* __restrict__ b_ih, const float* __restrict__ b_hh,
    const float* __restrict__ ph, const float* __restrict__ out_W,
    const float* __restrict__ out_b,
    float* __restrict__ out, float* __restrict__ h_new)
{
    __shared__ float red[256];
    const int b = blockIdx.x, h = threadIdx.x;
    const size_t gb = (size_t)b * G3;
    float ir = gi[gb + h]        + b_ih[h];
    float iz = gi[gb + 256 + h]  + b_ih[256 + h];
    float in = gi[gb + 512 + h]  + b_ih[512 + h];
    float hr = gh[gb + h]        + b_hh[h];
    float hz = gh[gb + 256 + h]  + b_hh[256 + h];
    float hn = gh[gb + 512 + h]  + b_hh[512 + h];
    float r = 1.f / (1.f + __expf(-(ir + hr)));
    float z = 1.f / (1.f + __expf(-(iz + hz)));
    float n = tanhf(in + r * hn);
    float hv = (1.f - z) * n + z * ph[(size_t)b * HD + h];
    h_new[(size_t)b * HD + h] = hv;

    red[h] = hv * out_W[h]; __syncthreads();
    for (int o = 128; o > 0; o >>= 1) {
        if (h < o) red[h] += red[h + o];
        __syncthreads();
    }
    if (h == 0) out[b] = red[0] + out_b[0];
}

// ---------------------------------------------------------------------------
extern "C" void kernel_launch(void* const* d_in, const int* in_sizes, int n_in,
                              void* d_out, int out_size, void* d_ws, size_t ws_size,
                              hipStream_t stream) {
    const float* enc    = (const float*)d_in[0];
    const float* ph     = (const float*)d_in[1];
    const float* y      = (const float*)d_in[2];
    const float* attn_W = (const float*)d_in[3];
    const float* attn_b = (const float*)d_in[4];
    const float* W_ih   = (const float*)d_in[5];
    const float* W_hh   = (const float*)d_in[6];
    const float* b_ih   = (const float*)d_in[7];
    const float* b_hh   = (const float*)d_in[8];
    const float* out_W  = (const float*)d_in[9];
    const float* out_b  = (const float*)d_in[10];

    const int B = in_sizes[1] / HD;   // 2048

    // workspace: logits/attn_w [B,512] | ctx [B,256] | gi [B,768] | gh [B,768]
    float* logits = (float*)d_ws;
    float* ctx    = logits + (size_t)B * SD;
    float* gi     = ctx    + (size_t)B * HD;
    float* gh     = gi     + (size_t)B * G3;

    float* out   = (float*)d_out;     // [B,1] first, then h_new [B,H]
    float* h_new = out + B;

    k_attn_logits<<<B / 16, 256, 0, stream>>>(ph, y, attn_W, logits);
    k_softmax    <<<B,      256, 0, stream>>>(logits, attn_b);
    k_ctx        <<<B,      256, 0, stream>>>(enc, logits, ctx);
    k_gru_gemm   <<<dim3(B / 16, G3 / 128), 256, 0, stream>>>(ctx, ph, W_ih, W_hh, gi, gh);
    k_gates      <<<B,      256, 0, stream>>>(gi, gh, b_ih, b_hh, ph, out_W, out_b, out, h_new);
}